// ClsbdCRF_1838246002800
// MI455X (gfx1250) — compile-verified
//
#include <hip/hip_runtime.h>
#include <hip/hip_bf16.h>

typedef float v2f __attribute__((ext_vector_type(2)));
typedef float v8f __attribute__((ext_vector_type(8)));

#define B_  4
#define C_  21
#define H_  128
#define W_  128
#define D_  5
#define EPSF 1e-5f
#define QS  34   // lQ row stride (64 rows x 32 channels, padded)
#define MS  66   // lM row stride (16 rows x 64 src pixels, padded -> conflict-free)

__global__ __launch_bounds__(32)
void clsbd_crf_kernel(const float* __restrict__ inp,
                      const float* __restrict__ feats,
                      const float* __restrict__ clsbd,
                      float* __restrict__ out)
{
    __shared__ float lQ[64 * QS];
    __shared__ float lMn[16 * MS];
    __shared__ float lMp[16 * MS];

    const int lane = threadIdx.x;
    const int t  = blockIdx.x;           // tile id: B * 32 * 32
    const int b  = t >> 10;
    const int ty = (t >> 5) & 31;
    const int tx = t & 31;
    const int H0 = ty * 4, W0 = tx * 4;

    // ---- zero weight matrices ----
    for (int i = lane; i < 16 * MS; i += 32) { lMn[i] = 0.f; lMp[i] = 0.f; }

    // ---- Phase A: Q[s][c] = inp * polarness over 8x8 padded source region ----
    const float invLogC = 1.0f / 3.0445224377234230f;   // 1/ln(21)
    const size_t HW = (size_t)H_ * W_;
    #pragma unroll
    for (int rep = 0; rep < 2; ++rep) {
        const int s  = lane + rep * 32;
        const int sy = s >> 3, sx = s & 7;
        const int gy = H0 - 2 + sy, gx = W0 - 2 + sx;
        const bool in = (gy >= 0) && (gy < H_) && (gx >= 0) && (gx < W_);
        if (in) {
            const float* p = inp + (size_t)b * C_ * HW + (size_t)gy * W_ + gx;
            float v[C_];
            float ent = 0.f;
            #pragma unroll
            for (int c = 0; c < C_; ++c) {
                float x = p[(size_t)c * HW];
                v[c] = x;
                ent -= x * __logf(x + EPSF);
            }
            const float pl = 1.0f - ent * invLogC;
            #pragma unroll
            for (int c = 0; c < C_; ++c) lQ[s * QS + c] = v[c] * pl;
        } else {
            #pragma unroll
            for (int c = 0; c < C_; ++c) lQ[s * QS + c] = 0.f;
        }
        #pragma unroll
        for (int c = C_; c < 32; ++c) lQ[s * QS + c] = 0.f;
    }

    // ---- Phase B: per-output-pixel 5x5 weights (lanes 0..15, one pixel each) ----
    if (lane < 16) {
        const int m  = lane;
        const int oy = m >> 2, ox = m & 3;
        const int py = H0 + oy, px = W0 + ox;
        const float* fb = feats + (size_t)b * D_ * HW;
        const float* cb = clsbd + (size_t)b * HW;

        float fc[D_];
        #pragma unroll
        for (int d = 0; d < D_; ++d) fc[d] = fb[(size_t)d * HW + (size_t)py * W_ + px];

        // zero-padded 5x5 boundary neighborhood
        float cl[5][5];
        #pragma unroll
        for (int a = 0; a < 5; ++a)
            #pragma unroll
            for (int d = 0; d < 5; ++d) {
                int gy = py + a - 2, gx = px + d - 2;
                cl[a][d] = (gy >= 0 && gy < H_ && gx >= 0 && gx < W_)
                           ? cb[(size_t)gy * W_ + gx] : 0.f;
            }

        // g2: ring-1 = value; ring-2 = max(value, expanded ring-1); center = 0
        float g2[5][5];
        g2[1][1] = cl[1][1]; g2[1][2] = cl[1][2]; g2[1][3] = cl[1][3];
        g2[2][1] = cl[2][1];                      g2[2][3] = cl[2][3];
        g2[3][1] = cl[3][1]; g2[3][2] = cl[3][2]; g2[3][3] = cl[3][3];
        g2[2][2] = 0.f;
        g2[0][0] = fmaxf(cl[0][0], cl[1][1]);
        g2[0][1] = fmaxf(cl[0][1], fmaxf(cl[1][1], cl[1][2]));
        g2[0][2] = fmaxf(cl[0][2], cl[1][2]);
        g2[0][3] = fmaxf(cl[0][3], fmaxf(cl[1][3], cl[1][2]));
        g2[0][4] = fmaxf(cl[0][4], cl[1][3]);
        g2[1][0] = fmaxf(cl[1][0], fmaxf(cl[1][1], cl[2][1]));
        g2[1][4] = fmaxf(cl[1][4], fmaxf(cl[1][3], cl[2][3]));
        g2[2][0] = fmaxf(cl[2][0], cl[2][1]);
        g2[2][4] = fmaxf(cl[2][4], cl[2][3]);
        g2[3][0] = fmaxf(cl[3][0], fmaxf(cl[3][1], cl[2][1]));
        g2[3][4] = fmaxf(cl[3][4], fmaxf(cl[3][3], cl[2][3]));
        g2[4][0] = fmaxf(cl[4][0], cl[3][1]);
        g2[4][1] = fmaxf(cl[4][1], fmaxf(cl[3][1], cl[3][2]));
        g2[4][2] = fmaxf(cl[4][2], cl[3][2]);
        g2[4][3] = fmaxf(cl[4][3], fmaxf(cl[3][3], cl[3][2]));
        g2[4][4] = fmaxf(cl[4][4], cl[3][3]);

        #pragma unroll
        for (int a = 0; a < 5; ++a)
            #pragma unroll
            for (int d = 0; d < 5; ++d) {
                int gy = py + a - 2, gx = px + d - 2;
                if (gy >= 0 && gy < H_ && gx >= 0 && gx < W_) {
                    float ss = 0.f;
                    #pragma unroll
                    for (int dd = 0; dd < D_; ++dd) {
                        float df = fc[dd] - fb[(size_t)dd * HW + (size_t)gy * W_ + gx];
                        ss += df * df;
                    }
                    float g1 = __expf(-0.5f * ss);
                    float gg = g2[a][d];
                    int   s  = (oy + a) * 8 + (ox + d);   // 8x8 padded source index
                    lMn[m * MS + s] = 10.0f * g1 - 5.0f * __logf(gg + EPSF);
                    lMp[m * MS + s] = -5.0f * __logf(1.0f - gg + EPSF);
                }
            }
    }
    __syncthreads();

    // ---- Phase C: (16x64)x(64x32) GEMM via V_WMMA_F32_16X16X4_F32 ----
    const int half = lane >> 4;   // K sub-block select per ISA A/B layout
    const int nr   = lane & 15;   // A: row M ; B: col N
    v8f accN0 = {}, accN1 = {}, accP0 = {}, accP1 = {};
    #pragma unroll
    for (int kk = 0; kk < 16; ++kk) {
        const int kb = 4 * kk + 2 * half;
        v2f aN, aP, b0, b1;
        aN.x = lMn[nr * MS + kb];      aN.y = lMn[nr * MS + kb + 1];
        aP.x = lMp[nr * MS + kb];      aP.y = lMp[nr * MS + kb + 1];
        b0.x = lQ[kb * QS + nr];       b0.y = lQ[(kb + 1) * QS + nr];
        b1.x = lQ[kb * QS + 16 + nr];  b1.y = lQ[(kb + 1) * QS + 16 + nr];
        accN0 = __builtin_amdgcn_wmma_f32_16x16x4_f32(false, aN, false, b0, (short)0, accN0, false, false);
        accN1 = __builtin_amdgcn_wmma_f32_16x16x4_f32(false, aN, false, b1, (short)0, accN1, false, false);
        accP0 = __builtin_amdgcn_wmma_f32_16x16x4_f32(false, aP, false, b0, (short)0, accP0, false, false);
        accP1 = __builtin_amdgcn_wmma_f32_16x16x4_f32(false, aP, false, b1, (short)0, accP1, false, false);
    }

    // ---- Phase D: scatter D-matrix (lane=N=channel, VGPR+8*half=M=pixel) ----
    const size_t outBase = (size_t)b * C_ * HW;
    const size_t plane   = (size_t)B_ * C_ * HW;   // msg_pos offset
    #pragma unroll
    for (int v = 0; v < 8; ++v) {
        const int m  = v + 8 * half;
        const int oy = m >> 2, ox = m & 3;
        const size_t pix = (size_t)(H0 + oy) * W_ + (W0 + ox);
        const int c0 = nr, c1 = nr + 16;
        out[outBase + (size_t)c0 * HW + pix]         = accN0[v];
        out[plane + outBase + (size_t)c0 * HW + pix] = accP0[v];
        if (c1 < C_) {
            out[outBase + (size_t)c1 * HW + pix]         = accN1[v];
            out[plane + outBase + (size_t)c1 * HW + pix] = accP1[v];
        }
    }
}

extern "C" void kernel_launch(void* const* d_in, const int* in_sizes, int n_in,
                              void* d_out, int out_size, void* d_ws, size_t ws_size,
                              hipStream_t stream) {
    (void)in_sizes; (void)n_in; (void)out_size; (void)d_ws; (void)ws_size;
    const float* inp   = (const float*)d_in[0];
    const float* feats = (const float*)d_in[1];
    const float* clsbd = (const float*)d_in[2];
    // d_in[3] = label, unused by the forward pass
    float* out = (float*)d_out;
    const int tiles = B_ * (H_ / 4) * (W_ / 4);   // 4096
    clsbd_crf_kernel<<<tiles, 32, 0, stream>>>(inp, feats, clsbd, out);
}